// DynResNet_54460185313346
// MI455X (gfx1250) — compile-verified
//
#include <hip/hip_runtime.h>
#include <math.h>

#define DIMX  64
#define KR    16
#define NL    9          // L+1 layers
#define DDIM  4096
#define BATCH 256
#define NCLS  10
#define HSTEP 1e-3f

typedef __attribute__((ext_vector_type(2))) float v2f;
typedef __attribute__((ext_vector_type(8))) float v8f;

// ---------------------------------------------------------------------------
// GEMM + bias + ReLU:  dY = relu(Xf @ W^T + b)
// M=256 (batch) x N=4096, K=4096.  Block = 256 threads (8 wave32), 64x64 tile.
// Wave w: rows 16*(w&3), cols 32*(w>>2); two 16x16 f32 accumulators per wave.
// Uses V_WMMA_F32_16X16X4_F32 (f32 in / f32 acc) to match reference numerics.
// ---------------------------------------------------------------------------
#define KB 32
#define LDSTR (KB + 4)   // pad k-stride to dodge LDS bank conflicts

__global__ __launch_bounds__(256)
void gemm_relu_kernel(const float* __restrict__ Xf,
                      const float* __restrict__ Wmat,
                      const float* __restrict__ bias,
                      float* __restrict__ dY)
{
    __shared__ float As[64][LDSTR];   // Xf tile  (m,k)
    __shared__ float Bs[64][LDSTR];   // W  tile  (n,k)  (B-frag = W^T)

    const int tid  = threadIdx.x;
    const int wave = tid >> 5;
    const int lane = tid & 31;
    const int half = lane >> 4;       // 0: lanes 0-15, 1: lanes 16-31
    const int l16  = lane & 15;

    const int wm = wave & 3;          // m sub-tile 0..3
    const int wn = wave >> 2;         // n strip  0..1

    const int m0 = blockIdx.y * 64;
    const int n0 = blockIdx.x * 64;

    v8f acc0 = {};
    v8f acc1 = {};

    for (int kt = 0; kt < DDIM; kt += KB) {
        for (int i = tid; i < 64 * KB; i += 256) {
            const int r = i >> 5;
            const int k = i & (KB - 1);
            As[r][k] = Xf  [(size_t)(m0 + r) * DDIM + kt + k];
            Bs[r][k] = Wmat[(size_t)(n0 + r) * DDIM + kt + k];
        }
        __syncthreads();
        #pragma unroll
        for (int kk = 0; kk < KB; kk += 4) {
            const int kl = kk + 2 * half;   // A: lanes0-15 hold K,K+1; 16-31 hold K+2,K+3
            v2f a, b0, b1;
            a.x  = As[wm * 16 + l16][kl];
            a.y  = As[wm * 16 + l16][kl + 1];
            b0.x = Bs[wn * 32 + l16][kl];          // B[k][n] = W[n][k]
            b0.y = Bs[wn * 32 + l16][kl + 1];
            b1.x = Bs[wn * 32 + 16 + l16][kl];
            b1.y = Bs[wn * 32 + 16 + l16][kl + 1];
            acc0 = __builtin_amdgcn_wmma_f32_16x16x4_f32(false, a, false, b0,
                                                         (short)0, acc0, false, false);
            acc1 = __builtin_amdgcn_wmma_f32_16x16x4_f32(false, a, false, b1,
                                                         (short)0, acc1, false, false);
        }
        __syncthreads();
    }

    // C/D layout: VGPR r -> (M=r, N=l16) lanes0-15 ; (M=r+8, N=l16) lanes16-31
    #pragma unroll
    for (int r = 0; r < 8; ++r) {
        const int row = m0 + wm * 16 + r + 8 * half;
        const int c0  = n0 + wn * 32 + l16;
        const int c1  = c0 + 16;
        float v0 = acc0[r] + (bias ? bias[c0] : 0.0f);
        float v1 = acc1[r] + (bias ? bias[c1] : 0.0f);
        dY[(size_t)row * DDIM + c0] = v0 > 0.0f ? v0 : 0.0f;
        dY[(size_t)row * DDIM + c1] = v1 > 0.0f ? v1 : 0.0f;
    }
}

// ---------------------------------------------------------------------------
// Init: parse X -> per-sample u (64x16), v (64x16, = vh^T), s (16x16);
// compute Xf0 = u @ s @ v^T.
// ---------------------------------------------------------------------------
__global__ __launch_bounds__(256)
void init_kernel(const float* __restrict__ X,
                 float* __restrict__ u_s, float* __restrict__ v_s,
                 float* __restrict__ s_s, float* __restrict__ Xf)
{
    const int b   = blockIdx.x;
    const int tid = threadIdx.x;
    const float* X0 = X + (size_t)b * 3 * DDIM;       // u flat (64x16)
    const float* X1 = X0 + DDIM;                      // s_block flat (64x16)
    const float* X2 = X1 + DDIM;                      // vh flat (16x64)

    __shared__ float Lu[DIMX][KR], Lv[DIMX][KR], Ls[KR][KR], US[DIMX][KR];

    for (int i = tid; i < DIMX * KR; i += 256) {
        const int d = i >> 4, k = i & 15;
        Lu[d][k] = X0[i];
        Lv[d][k] = X2[k * DIMX + d];                  // v = vh^T
    }
    for (int i = tid; i < KR * KR; i += 256)
        Ls[i >> 4][i & 15] = X1[i];                   // top-left 16x16 of s_block
    __syncthreads();

    for (int i = tid; i < DIMX * KR; i += 256) {
        const int r = i >> 4, c = i & 15;
        float acc = 0.0f;
        for (int k = 0; k < KR; ++k) acc += Lu[r][k] * Ls[k][c];
        US[r][c] = acc;
    }
    __syncthreads();

    float* ub = u_s + (size_t)b * DIMX * KR;
    float* vb = v_s + (size_t)b * DIMX * KR;
    float* sb = s_s + (size_t)b * KR * KR;
    for (int i = tid; i < DIMX * KR; i += 256) { ub[i] = Lu[i >> 4][i & 15]; vb[i] = Lv[i >> 4][i & 15]; }
    for (int i = tid; i < KR * KR;  i += 256) sb[i] = Ls[i >> 4][i & 15];
    for (int i = tid; i < DDIM; i += 256) {
        const int r = i >> 6, c = i & 63;
        float acc = 0.0f;
        for (int k = 0; k < KR; ++k) acc += US[r][k] * Lv[c][k];
        Xf[(size_t)b * DDIM + i] = acc;
    }
}

// ---------------------------------------------------------------------------
// Cayley retraction on one factor (u or v), via Woodbury (16x16 inverse):
//   F = (I - L L^T) dL   (dL passed in D, overwritten by F)
//   G = H L^T F ; T = L^T L ; S = I + 0.5 G^2
//   L <- L - H * F @ ( G @ (T - 0.5 G S^-1 G T) )
// ---------------------------------------------------------------------------
__device__ __forceinline__ void cayley_update(
    float (*Lm)[KR], float (*D)[KR],
    float (*G)[KR], float (*T)[KR], float (*GT)[KR],
    float (*Z)[KR], float (*Y)[KR], float (*W16)[KR],
    float (*AUG)[2 * KR], float* fcol, int tid)
{
    // M1 = L^T D   (stash in G temporarily)
    for (int i = tid; i < KR * KR; i += 256) {
        const int k = i >> 4, j = i & 15;
        float acc = 0.0f;
        for (int r = 0; r < DIMX; ++r) acc += Lm[r][k] * D[r][j];
        G[k][j] = acc;
    }
    __syncthreads();
    // F = D - L @ M1  (in place)
    for (int i = tid; i < DIMX * KR; i += 256) {
        const int r = i >> 4, j = i & 15;
        float acc = D[r][j];
        for (int k = 0; k < KR; ++k) acc -= Lm[r][k] * G[k][j];
        D[r][j] = acc;
    }
    __syncthreads();
    // G = H * L^T F ;  T = L^T L
    for (int i = tid; i < 2 * KR * KR; i += 256) {
        const int w = (i >= KR * KR) ? 1 : 0;
        const int j2 = w ? i - KR * KR : i;
        const int k = j2 >> 4, j = j2 & 15;
        float acc = 0.0f;
        if (!w) { for (int r = 0; r < DIMX; ++r) acc += Lm[r][k] * D[r][j];  G[k][j] = HSTEP * acc; }
        else    { for (int r = 0; r < DIMX; ++r) acc += Lm[r][k] * Lm[r][j]; T[k][j] = acc; }
    }
    __syncthreads();
    // GT = G@T ;  AUG = [ I + 0.5 G@G | I ]
    for (int i = tid; i < 2 * KR * KR; i += 256) {
        const int w = (i >= KR * KR) ? 1 : 0;
        const int j2 = w ? i - KR * KR : i;
        const int k = j2 >> 4, j = j2 & 15;
        float acc = 0.0f;
        if (!w) { for (int q = 0; q < KR; ++q) acc += G[k][q] * T[q][j]; GT[k][j] = acc; }
        else {
            for (int q = 0; q < KR; ++q) acc += G[k][q] * G[q][j];
            AUG[k][j]      = 0.5f * acc + ((k == j) ? 1.0f : 0.0f);
            AUG[k][j + KR] = (k == j) ? 1.0f : 0.0f;
        }
    }
    __syncthreads();
    // Gauss-Jordan (S = I + O(H^2): no pivoting needed)
    for (int p = 0; p < KR; ++p) {
        const float pivinv = 1.0f / AUG[p][p];
        __syncthreads();
        if (tid < 2 * KR) AUG[p][tid] *= pivinv;
        __syncthreads();
        if (tid < KR) fcol[tid] = (tid == p) ? 0.0f : AUG[tid][p];
        __syncthreads();
        for (int i = tid; i < KR * 2 * KR; i += 256) {
            const int r = i >> 5, c = i & 31;
            if (r != p) AUG[r][c] -= fcol[r] * AUG[p][c];
        }
        __syncthreads();
    }
    // Z = S^-1 @ GT ; Y = T - 0.5 G@Z ; W16 = G@Y
    for (int i = tid; i < KR * KR; i += 256) {
        const int k = i >> 4, j = i & 15;
        float acc = 0.0f;
        for (int q = 0; q < KR; ++q) acc += AUG[k][KR + q] * GT[q][j];
        Z[k][j] = acc;
    }
    __syncthreads();
    for (int i = tid; i < KR * KR; i += 256) {
        const int k = i >> 4, j = i & 15;
        float acc = 0.0f;
        for (int q = 0; q < KR; ++q) acc += G[k][q] * Z[q][j];
        Y[k][j] = T[k][j] - 0.5f * acc;
    }
    __syncthreads();
    for (int i = tid; i < KR * KR; i += 256) {
        const int k = i >> 4, j = i & 15;
        float acc = 0.0f;
        for (int q = 0; q < KR; ++q) acc += G[k][q] * Y[q][j];
        W16[k][j] = acc;
    }
    __syncthreads();
    // L <- L - H * F @ W16
    for (int i = tid; i < DIMX * KR; i += 256) {
        const int r = i >> 4, j = i & 15;
        float acc = 0.0f;
        for (int k = 0; k < KR; ++k) acc += D[r][k] * W16[k][j];
        Lm[r][j] -= HSTEP * acc;
    }
    __syncthreads();
}

// ---------------------------------------------------------------------------
// Per-sample update: grads, Cayley u/v updates, s += H dS, new Xf = u s v^T,
// and write X_transformed[:, :, layer].
// ---------------------------------------------------------------------------
__global__ __launch_bounds__(256)
void update_kernel(const float* __restrict__ dY_g,
                   float* __restrict__ u_s, float* __restrict__ v_s,
                   float* __restrict__ s_s, float* __restrict__ Xf,
                   float* __restrict__ out_tr, int layer)
{
    const int b   = blockIdx.x;
    const int tid = threadIdx.x;

    __shared__ float Lu[DIMX][KR], Lv[DIMX][KR], Ls[KR][KR];
    __shared__ float LdY[DIMX][DIMX];
    __shared__ float D1[DIMX][KR], D2[DIMX][KR];   // P/dU/Fu and Q/dV/Fv
    __shared__ float dS[KR][KR];
    __shared__ float G[KR][KR], T[KR][KR], GT[KR][KR];
    __shared__ float Z[KR][KR], Y[KR][KR], W16[KR][KR];
    __shared__ float AUG[KR][2 * KR];
    __shared__ float fcol[KR];

    float* ub = u_s + (size_t)b * DIMX * KR;
    float* vb = v_s + (size_t)b * DIMX * KR;
    float* sb = s_s + (size_t)b * KR * KR;
    const float* dyb = dY_g + (size_t)b * DDIM;

    for (int i = tid; i < DIMX * KR; i += 256) { Lu[i >> 4][i & 15] = ub[i]; Lv[i >> 4][i & 15] = vb[i]; }
    for (int i = tid; i < KR * KR;  i += 256) Ls[i >> 4][i & 15] = sb[i];
    for (int i = tid; i < DDIM;     i += 256) LdY[i >> 6][i & 63] = dyb[i];
    __syncthreads();

    // P = dY@v (-> D1) ; Q = dY^T@u (-> D2)
    for (int i = tid; i < 2 * DIMX * KR; i += 256) {
        const int w = (i >= DIMX * KR) ? 1 : 0;
        const int j2 = w ? i - DIMX * KR : i;
        const int r = j2 >> 4, c = j2 & 15;
        float acc = 0.0f;
        if (!w) { for (int k = 0; k < DIMX; ++k) acc += LdY[r][k] * Lv[k][c]; D1[r][c] = acc; }
        else    { for (int k = 0; k < DIMX; ++k) acc += LdY[k][r] * Lu[k][c]; D2[r][c] = acc; }
    }
    __syncthreads();
    // dS = u^T dY v = Q^T @ v
    for (int i = tid; i < KR * KR; i += 256) {
        const int k = i >> 4, l = i & 15;
        float acc = 0.0f;
        for (int j = 0; j < DIMX; ++j) acc += D2[j][k] * Lv[j][l];
        dS[k][l] = acc;
    }
    __syncthreads();
    // dU = P * s_inv ; dV = Q * s_inv   (diagonal inverse of OLD s)
    for (int i = tid; i < DIMX * KR; i += 256) {
        const int r = i >> 4, c = i & 15;
        const float inv = 1.0f / Ls[c][c];
        D1[r][c] *= inv;
        D2[r][c] *= inv;
    }
    __syncthreads();

    cayley_update(Lu, D1, G, T, GT, Z, Y, W16, AUG, fcol, tid);   // u <- cay(...) u
    cayley_update(Lv, D2, G, T, GT, Z, Y, W16, AUG, fcol, tid);   // v <- cay(...) v

    for (int i = tid; i < KR * KR; i += 256) Ls[i >> 4][i & 15] += HSTEP * dS[i >> 4][i & 15];
    __syncthreads();
    // US = u@s  (reuse D1)
    for (int i = tid; i < DIMX * KR; i += 256) {
        const int r = i >> 4, c = i & 15;
        float acc = 0.0f;
        for (int k = 0; k < KR; ++k) acc += Lu[r][k] * Ls[k][c];
        D1[r][c] = acc;
    }
    __syncthreads();
    // Xf = US @ v^T ; emit transformed slice; persist state
    for (int i = tid; i < DDIM; i += 256) {
        const int r = i >> 6, c = i & 63;
        float acc = 0.0f;
        for (int k = 0; k < KR; ++k) acc += D1[r][k] * Lv[c][k];
        Xf[(size_t)b * DDIM + i] = acc;
        out_tr[(size_t)b * DDIM * NL + (size_t)i * NL + layer] = acc;
    }
    for (int i = tid; i < DIMX * KR; i += 256) { ub[i] = Lu[i >> 4][i & 15]; vb[i] = Lv[i >> 4][i & 15]; }
    for (int i = tid; i < KR * KR;  i += 256) sb[i] = Ls[i >> 4][i & 15];
}

// ---------------------------------------------------------------------------
// Classifier + softmax
// ---------------------------------------------------------------------------
__global__ __launch_bounds__(256)
void classify_kernel(const float* __restrict__ Xf,
                     const float* __restrict__ Wc,
                     const float* __restrict__ bc,
                     float* __restrict__ pred,
                     float* __restrict__ cls)
{
    const int b   = blockIdx.x;
    const int tid = threadIdx.x;
    __shared__ float red[256];
    __shared__ float logit[NCLS];
    const float* x = Xf + (size_t)b * DDIM;

    for (int c = 0; c < NCLS; ++c) {
        float acc = 0.0f;
        for (int d = tid; d < DDIM; d += 256) acc += x[d] * Wc[(size_t)c * DDIM + d];
        red[tid] = acc;
        __syncthreads();
        for (int s = 128; s > 0; s >>= 1) {
            if (tid < s) red[tid] += red[tid + s];
            __syncthreads();
        }
        if (tid == 0) logit[c] = red[0] + bc[c];
        __syncthreads();
    }
    if (tid == 0) {
        float mx = logit[0];
        for (int c = 1; c < NCLS; ++c) mx = fmaxf(mx, logit[c]);
        float e[NCLS], sum = 0.0f;
        for (int c = 0; c < NCLS; ++c) { e[c] = expf(logit[c] - mx); sum += e[c]; }
        for (int c = 0; c < NCLS; ++c) {
            cls [b * NCLS + c] = logit[c];
            pred[b * NCLS + c] = e[c] / sum;
        }
    }
}

// ---------------------------------------------------------------------------
extern "C" void kernel_launch(void* const* d_in, const int* in_sizes, int n_in,
                              void* d_out, int out_size, void* d_ws, size_t ws_size,
                              hipStream_t stream)
{
    (void)in_sizes; (void)n_in; (void)out_size; (void)ws_size;
    const float* X  = (const float*)d_in[0];
    const float* W0 = (const float*)d_in[1];
    const float* W  = (const float*)d_in[2];
    const float* bb = (const float*)d_in[3];
    const float* Wc = (const float*)d_in[4];
    const float* bc = (const float*)d_in[5];

    float* out      = (float*)d_out;
    float* out_pred = out;
    float* out_cls  = out + BATCH * NCLS;
    float* out_tr   = out + 2 * BATCH * NCLS;

    float* ws  = (float*)d_ws;
    float* Xf  = ws;                               // BATCH*DDIM
    float* dY  = Xf  + (size_t)BATCH * DDIM;       // BATCH*DDIM
    float* u_s = dY  + (size_t)BATCH * DDIM;       // BATCH*64*16
    float* v_s = u_s + (size_t)BATCH * DIMX * KR;  // BATCH*64*16
    float* s_s = v_s + (size_t)BATCH * DIMX * KR;  // BATCH*16*16

    init_kernel<<<BATCH, 256, 0, stream>>>(X, u_s, v_s, s_s, Xf);
    for (int layer = 0; layer < NL; ++layer) {
        const float* Wl = (layer == 0) ? W0 : (W  + (size_t)(layer - 1) * DDIM * DDIM);
        const float* bl = (layer == 0) ? nullptr : (bb + (size_t)(layer - 1) * DDIM);
        gemm_relu_kernel<<<dim3(DDIM / 64, BATCH / 64), 256, 0, stream>>>(Xf, Wl, bl, dY);
        update_kernel<<<BATCH, 256, 0, stream>>>(dY, u_s, v_s, s_s, Xf, out_tr, layer);
    }
    classify_kernel<<<BATCH, 256, 0, stream>>>(Xf, Wc, bc, out_pred, out_cls);
}